// RoutingLayer_2173253452540
// MI455X (gfx1250) — compile-verified
//
#include <hip/hip_runtime.h>
#include <math.h>

// CDNA5 / gfx1250. wave32. K=8 capsules, DD=16, D=128, CUT=5.
#define D_   128
#define KC   8
#define DD_  16
#define CUT_ 5
#define M_   32

typedef float v2f __attribute__((ext_vector_type(2)));
typedef float v8f __attribute__((ext_vector_type(8)));

// ---------------------------------------------------------------------------
// Kernel A: per-capsule L2 normalize. One 128-thread block per row; block N
// writes the zero pad row. 16-lane sub-group reduction for the capsule norm.
// ---------------------------------------------------------------------------
__global__ __launch_bounds__(128) void norm_kernel(const float* __restrict__ x,
                                                   float* __restrict__ xn, int N) {
  const int node = blockIdx.x;
  const int t = threadIdx.x;
  if (node >= N) { xn[(size_t)node * D_ + t] = 0.0f; return; }
  float v = x[(size_t)node * D_ + t];
  float s = v * v;
  // sum of squares within each 16-lane capsule group (wave32: 2 capsules/wave)
  s += __shfl_xor(s, 1, 16);
  s += __shfl_xor(s, 2, 16);
  s += __shfl_xor(s, 4, 16);
  s += __shfl_xor(s, 8, 16);
  float nrm = fmaxf(sqrtf(s), 1e-12f);
  xn[(size_t)node * D_ + t] = v / nrm;
}

// ---------------------------------------------------------------------------
// Kernel B: routing. One wave32 per node. z (32x128 f32 = 16KB) staged in LDS
// once; 6 routing iterations entirely on-chip. Both einsums run on
// V_WMMA_F32_16X16X4_F32 with the vector operand broadcast across B columns.
// ---------------------------------------------------------------------------
__global__ __launch_bounds__(32) void routing_kernel(const float* __restrict__ xn,
                                                     const int* __restrict__ nbr,
                                                     const int* __restrict__ p_maxiter,
                                                     float* __restrict__ u_out,
                                                     int* __restrict__ selcap,
                                                     int N, int M) {
  __shared__ float zs[M_ * D_];   // [m][k*16+c]
  __shared__ float ps[M_ * KC];   // [m][k]
  __shared__ float us[D_];        // [k*16+c]
  __shared__ float xs[D_];        // normalized x of this node

  const int node = blockIdx.x;
  const int lane = threadIdx.x;          // 0..31
  const int half = lane >> 4;            // 0/1
  const int lo   = lane & 15;
  const int T = p_maxiter[0];

  // stage own (normalized) row
  ((float4*)xs)[lane] = ((const float4*)(xn + (size_t)node * D_))[lane];

  // gather 32 neighbor rows -> LDS (coalesced float4 per lane per row)
  int gl = nbr[node * M + lane];
  for (int r = 0; r < M_; ++r) {
    int g = __shfl(gl, r, 32);
    int row = (g < 0) ? N : g;                 // wrap(-1) -> zero row
    ((float4*)(zs + r * D_))[lane] =
        ((const float4*)(xn + (size_t)row * D_))[lane];
  }
  __syncthreads();

  for (int it = 0; it < T; ++it) {
    if (it == 0) {
      // softmax(zeros) = 1/8
      #pragma unroll
      for (int q = 0; q < KC; ++q) ps[lane * KC + q] = 0.125f;
    } else {
      // p_raw[m][k] = sum_c z[m,k,c] * u[k,c]   (per capsule, M=32 -> 2 tiles)
      for (int k = 0; k < KC; ++k) {
        v8f d0 = {}; v8f d1 = {};
        #pragma unroll
        for (int s = 0; s < 4; ++s) {
          const int c0 = 4 * s + 2 * half;     // K-slot pair for this lane-half
          v2f b; b.x = us[k * DD_ + c0];     b.y = us[k * DD_ + c0 + 1];
          v2f a0; a0.x = zs[lo * D_        + k * DD_ + c0];
                  a0.y = zs[lo * D_        + k * DD_ + c0 + 1];
          v2f a1; a1.x = zs[(16 + lo) * D_ + k * DD_ + c0];
                  a1.y = zs[(16 + lo) * D_ + k * DD_ + c0 + 1];
          d0 = __builtin_amdgcn_wmma_f32_16x16x4_f32(false, a0, false, b, (short)0, d0, false, false);
          d1 = __builtin_amdgcn_wmma_f32_16x16x4_f32(false, a1, false, b, (short)0, d1, false, false);
        }
        // all D columns equal -> lane (lo==0) of each half owns rows 8*half+v
        if (lo == 0) {
          #pragma unroll
          for (int v = 0; v < 8; ++v) {
            ps[(8 * half + v) * KC + k]      = d0[v];
            ps[(16 + 8 * half + v) * KC + k] = d1[v];
          }
        }
      }
      __syncthreads();
      // softmax over k: one m-row per lane
      float pv[KC];
      float mx = -3.0e38f;
      #pragma unroll
      for (int q = 0; q < KC; ++q) { pv[q] = ps[lane * KC + q]; mx = fmaxf(mx, pv[q]); }
      float ssum = 0.0f;
      #pragma unroll
      for (int q = 0; q < KC; ++q) { pv[q] = expf(pv[q] - mx); ssum += pv[q]; }
      float inv = 1.0f / ssum;
      #pragma unroll
      for (int q = 0; q < KC; ++q) ps[lane * KC + q] = pv[q] * inv;
    }
    __syncthreads();

    // u[k,c] = sum_m z[m,k,c] * p[m,k] + x[k,c]  ; K-dim = m (32) -> 8 steps
    for (int k = 0; k < KC; ++k) {
      v8f d = {};
      #pragma unroll
      for (int s = 0; s < 8; ++s) {
        const int m0 = 4 * s + 2 * half;
        v2f a; a.x = zs[m0 * D_ + k * DD_ + lo];
               a.y = zs[(m0 + 1) * D_ + k * DD_ + lo];
        v2f b; b.x = ps[m0 * KC + k];
               b.y = ps[(m0 + 1) * KC + k];
        d = __builtin_amdgcn_wmma_f32_16x16x4_f32(false, a, false, b, (short)0, d, false, false);
      }
      float uv[8];
      #pragma unroll
      for (int v = 0; v < 8; ++v) uv[v] = d[v] + xs[k * DD_ + 8 * half + v];
      if (it < T - 1) {
        float s2 = 0.0f;
        #pragma unroll
        for (int v = 0; v < 8; ++v) s2 += uv[v] * uv[v];
        s2 += __shfl_xor(s2, 16, 32);          // combine the two halves
        float inv = 1.0f / fmaxf(sqrtf(s2), 1e-12f);
        #pragma unroll
        for (int v = 0; v < 8; ++v) uv[v] *= inv;
      }
      if (lo == 0) {
        #pragma unroll
        for (int v = 0; v < 8; ++v) us[k * DD_ + 8 * half + v] = uv[v];
      }
    }
    __syncthreads();
  }

  // write final (unnormalized) u, coalesced
  ((float4*)(u_out + (size_t)node * D_))[lane] = ((float4*)us)[lane];

  // argmax capsule (first-max wins, as jnp.argmax) for the first CUT slots
  if (lane < CUT_) {
    int best = 0;
    float bv = ps[lane * KC + 0];
    #pragma unroll
    for (int q = 1; q < KC; ++q) {
      float v = ps[lane * KC + q];
      if (v > bv) { bv = v; best = q; }
    }
    selcap[node * CUT_ + lane] = best;
  }
}

// ---------------------------------------------------------------------------
// Kernel C: meta (two-hop gather). thread t -> (i = t/16, c = t%16).
// B[i,node,s] = nb if (selcap==i && nb>=0) else -1; -1 hops hit the zero row,
// so they contribute exactly 0 and are skipped.
// ---------------------------------------------------------------------------
__global__ __launch_bounds__(128) void meta_kernel(const float* __restrict__ xn,
                                                   const int* __restrict__ nbr,
                                                   const int* __restrict__ selcap,
                                                   const int* __restrict__ p_last,
                                                   const float* __restrict__ u_out,
                                                   float* __restrict__ meta_out,
                                                   int N, int M) {
  if (p_last[0] == 0) return;
  const int node = blockIdx.x;
  const int t = threadIdx.x;
  const int i = t >> 4;
  const int c = t & 15;

  __shared__ int nb1[CUT_], cap1[CUT_];
  if (t < CUT_) {
    nb1[t]  = nbr[node * M + t];
    cap1[t] = selcap[node * CUT_ + t];
  }
  __syncthreads();

  float acc = 0.0f;
  for (int j = 0; j < KC; ++j) {
    float aj = 0.0f;
    #pragma unroll
    for (int s = 0; s < CUT_; ++s) {
      int g1 = nb1[s];
      if (g1 >= 0 && cap1[s] == i) {
        #pragma unroll
        for (int s2 = 0; s2 < CUT_; ++s2) {
          int g2 = nbr[g1 * M + s2];
          if (g2 >= 0 && selcap[g1 * CUT_ + s2] == j) {
            aj += xn[(size_t)g2 * D_ + j * DD_ + c];
          }
        }
      }
    }
    acc += aj / 25.0f;                          // mean over CUT*CUT, then sum_j
  }
  meta_out[(size_t)node * D_ + t] = u_out[(size_t)node * D_ + t] + acc;
}

// ---------------------------------------------------------------------------
extern "C" void kernel_launch(void* const* d_in, const int* in_sizes, int n_in,
                              void* d_out, int out_size, void* d_ws, size_t ws_size,
                              hipStream_t stream) {
  const float* x       = (const float*)d_in[0];
  const int*   nbr     = (const int*)d_in[1];
  const int*   maxiter = (const int*)d_in[2];
  const int*   last    = (const int*)d_in[3];

  const int N = in_sizes[0] / D_;
  const int M = in_sizes[1] / N;

  float* u_out    = (float*)d_out;                         // N*128
  float* meta_out = (float*)d_out + (size_t)N * D_;        // N*128

  // workspace: xn[(N+1)*128] f32, then selcap[N*CUT] i32
  float* xn = (float*)d_ws;
  size_t xn_bytes = (size_t)(N + 1) * D_ * sizeof(float);
  int* selcap = (int*)((char*)d_ws + xn_bytes);

  norm_kernel<<<N + 1, 128, 0, stream>>>(x, xn, N);
  routing_kernel<<<N, 32, 0, stream>>>(xn, nbr, maxiter, u_out, selcap, N, M);
  meta_kernel<<<N, 128, 0, stream>>>(xn, nbr, selcap, last, u_out, meta_out, N, M);
}